// DiceLoss_44538810859601
// MI455X (gfx1250) — compile-verified
//
#include <hip/hip_runtime.h>

// ---------------------------------------------------------------------------
// DiceLoss (binary, soft, average='batch', channel_reduction='mean')
// input/target: fp32 [16, 3, 512, 512]; output: fp32 scalar.
//
// Bandwidth-bound: 100.7 MB of reads -> ~4.3us floor at 23.3 TB/s.
// Strategy: streaming b128 non-temporal loads, per-thread f32 accumulation,
// WMMA-based intra-wave allreduce (V_WMMA_F32_16X16X4_F32 with B=ones),
// deterministic two-pass final reduction (no float atomics).
// ---------------------------------------------------------------------------

typedef float v2f __attribute__((ext_vector_type(2)));
typedef float v4f __attribute__((ext_vector_type(4)));
typedef float v8f __attribute__((ext_vector_type(8)));

#define NB       16
#define NC       3
#define NSP      (512 * 512)          // 262144 spatial elements per plane
#define NPLANES  (NB * NC)            // 48 planes
#define SPLITS   32                   // blocks per plane
#define NBLOCKS  (NPLANES * SPLITS)   // 1536 blocks -> 12288 waves
#define THREADS  256                  // 8 waves per block (wave32)
#define CHUNK    (NSP / SPLITS)       // 8192 floats per block per tensor
#define CHUNK4   (CHUNK / 4)          // 2048 float4 per block per tensor
#define ITERS    (CHUNK4 / THREADS)   // 8 float4 loads per thread per tensor
#define EPSV     1e-5

// Full 32-lane allreduce of one f32 value per lane, on the matrix engine.
// Stage 1: D = A x Ones with A=(s,0) per lane:
//   A[m][0]=s[m], A[m][1]=0, A[m][2]=s[m+16], A[m][3]=0  (ISA 16x4 A layout)
//   => D[m][n] = s[m] + s[m+16] = P[m]  (all n)
//   D VGPR j holds P[j] in lanes 0-15 and P[j+8] in lanes 16-31.
// Stage 2: four accumulating WMMAs with A built from (d[2i], d[2i+1]) sum all
// 16 pair-sums P[0..15]; every lane of e[0] ends with the full 32-lane total.
// B = all-ones makes the result independent of the (undocumented) B layout.
__device__ __forceinline__ float wave_allreduce_f32(float s) {
    v2f ones; ones[0] = 1.0f; ones[1] = 1.0f;
    v2f a;    a[0] = s;       a[1] = 0.0f;
    v8f c0 = {};
    v8f d = __builtin_amdgcn_wmma_f32_16x16x4_f32(
        false, a, false, ones, (short)0, c0, false, false);

    v8f e = {};
    v2f p;
    p[0] = d[0]; p[1] = d[1];
    e = __builtin_amdgcn_wmma_f32_16x16x4_f32(
        false, p, false, ones, (short)0, e, false, false);
    p[0] = d[2]; p[1] = d[3];
    e = __builtin_amdgcn_wmma_f32_16x16x4_f32(
        false, p, false, ones, (short)0, e, false, false);
    p[0] = d[4]; p[1] = d[5];
    e = __builtin_amdgcn_wmma_f32_16x16x4_f32(
        false, p, false, ones, (short)0, e, false, false);
    p[0] = d[6]; p[1] = d[7];
    e = __builtin_amdgcn_wmma_f32_16x16x4_f32(
        false, p, false, ones, (short)0, e, false, false);
    return e[0];
}

__global__ __launch_bounds__(THREADS)
void dice_partial_kernel(const float* __restrict__ X,
                         const float* __restrict__ T,
                         float* __restrict__ ws) {
    const int blk   = blockIdx.x;
    const int plane = blk >> 5;        // / SPLITS
    const int split = blk & (SPLITS - 1);
    const int base  = plane * NSP + split * CHUNK;   // 16B-aligned (8192-float mult.)

    const v4f* __restrict__ x4 = (const v4f*)(X + base);
    const v4f* __restrict__ t4 = (const v4f*)(T + base);

    float sxt = 0.0f, sxx = 0.0f, stt = 0.0f;
#pragma unroll
    for (int it = 0; it < ITERS; ++it) {
        const int i = it * THREADS + threadIdx.x;
        v4f xv = __builtin_nontemporal_load(&x4[i]);   // global_load_b128, TH=NT
        v4f tv = __builtin_nontemporal_load(&t4[i]);
        sxt = fmaf(xv[0], tv[0], sxt); sxt = fmaf(xv[1], tv[1], sxt);
        sxt = fmaf(xv[2], tv[2], sxt); sxt = fmaf(xv[3], tv[3], sxt);
        sxx = fmaf(xv[0], xv[0], sxx); sxx = fmaf(xv[1], xv[1], sxx);
        sxx = fmaf(xv[2], xv[2], sxx); sxx = fmaf(xv[3], xv[3], sxx);
        stt = fmaf(tv[0], tv[0], stt); stt = fmaf(tv[1], tv[1], stt);
        stt = fmaf(tv[2], tv[2], stt); stt = fmaf(tv[3], tv[3], stt);
    }

    // Intra-wave allreduce on the matrix engine (EXEC is all-ones here:
    // full 256-thread block, uniform control flow).
    const float wxt  = wave_allreduce_f32(sxt);
    const float wden = wave_allreduce_f32(sxx + stt);

    __shared__ float red[THREADS / 32][2];
    const int wave = threadIdx.x >> 5;
    const int lane = threadIdx.x & 31;
    if (lane == 0) { red[wave][0] = wxt; red[wave][1] = wden; }
    __syncthreads();

    if (threadIdx.x == 0) {
        float xt = 0.0f, den = 0.0f;
#pragma unroll
        for (int w = 0; w < THREADS / 32; ++w) {
            xt  += red[w][0];
            den += red[w][1];
        }
        ws[2 * blk + 0] = xt;    // overwritten every call: no init pass needed
        ws[2 * blk + 1] = den;
    }
}

__global__ void dice_final_kernel(const float* __restrict__ ws,
                                  float* __restrict__ out) {
    if (threadIdx.x == 0 && blockIdx.x == 0) {
        double dice_sum = 0.0;
        for (int c = 0; c < NC; ++c) {
            double xt = 0.0, den = 0.0;
            for (int p = c; p < NPLANES; p += NC) {       // planes of channel c
                const int b0 = p * SPLITS;
                for (int s = 0; s < SPLITS; ++s) {
                    xt  += (double)ws[2 * (b0 + s) + 0];
                    den += (double)ws[2 * (b0 + s) + 1];
                }
            }
            dice_sum += (2.0 * xt + EPSV) / (den + EPSV);
        }
        out[0] = (float)(1.0 - dice_sum / (double)NC);
    }
}

extern "C" void kernel_launch(void* const* d_in, const int* in_sizes, int n_in,
                              void* d_out, int out_size, void* d_ws, size_t ws_size,
                              hipStream_t stream) {
    (void)in_sizes; (void)n_in; (void)out_size; (void)ws_size;
    const float* X = (const float*)d_in[0];
    const float* T = (const float*)d_in[1];
    float* ws  = (float*)d_ws;    // needs 1536 * 2 * 4 = 12288 bytes
    float* out = (float*)d_out;

    dice_partial_kernel<<<dim3(NBLOCKS), dim3(THREADS), 0, stream>>>(X, T, ws);
    dice_final_kernel<<<dim3(1), dim3(1), 0, stream>>>(ws, out);
}